// GGD_1614907703322
// MI455X (gfx1250) — compile-verified
//
#include <hip/hip_runtime.h>
#include <hip/hip_bf16.h>
#include <math.h>

typedef __attribute__((ext_vector_type(2))) float v2f;
typedef __attribute__((ext_vector_type(8))) float v8f;

#define N_NODES 100000
#define N_EDGES 1600000
#define FD      128   // feature dim == hidden dim

// ---------------------------------------------------------------------------
// small setup kernels
// ---------------------------------------------------------------------------
__global__ void k_zero(float* __restrict__ deg, float* __restrict__ s,
                       float* __restrict__ sview, int n) {
    int i = blockIdx.x * blockDim.x + threadIdx.x;
    if (i < n) { deg[i] = 0.0f; s[i] = 0.0f; }
    if (i == 0) { sview[0] = 0.0f; sview[1] = 0.0f; }
}

__global__ void k_deg(const int* __restrict__ dst, float* __restrict__ deg, int E) {
    int e = blockIdx.x * blockDim.x + threadIdx.x;
    if (e < E) atomicAdd(&deg[dst[e]], 1.0f);
}

__global__ void k_norm(float* __restrict__ norm, int n) {
    int i = blockIdx.x * blockDim.x + threadIdx.x;
    if (i < n) norm[i] = rsqrtf(norm[i] + 1.0f);
}

// S[n] = sum over edges with src==n of norm[dst]
__global__ void k_srcsum(const int* __restrict__ src, const int* __restrict__ dst,
                         const float* __restrict__ norm, float* __restrict__ S, int E) {
    int e = blockIdx.x * blockDim.x + threadIdx.x;
    if (e < E) atomicAdd(&S[src[e]], norm[dst[e]]);
}

// c[n] = norm[n] * (norm[n] + S[n])
__global__ void k_coeff(const float* __restrict__ norm, float* __restrict__ c, int n) {
    int i = blockIdx.x * blockDim.x + threadIdx.x;
    if (i < n) c[i] = norm[i] * (norm[i] + c[i]);
}

// v = W2 @ (Wp @ 1) ; konst = N * (b2.(Wp@1) + sum(bp))
__global__ void k_proj(const float* __restrict__ W2, const float* __restrict__ b2,
                       const float* __restrict__ Wp, const float* __restrict__ bp,
                       float* __restrict__ v, float* __restrict__ konst) {
    __shared__ float w[FD];
    __shared__ float red[FD];
    int k = threadIdx.x;
    float s = 0.0f;
    for (int j = 0; j < FD; ++j) s += Wp[k * FD + j];     // row-sum of Wp
    w[k] = s;
    __syncthreads();
    float acc = 0.0f;
    for (int j = 0; j < FD; ++j) acc += W2[k * FD + j] * w[j];
    v[k] = acc;
    red[k] = b2[k] * w[k] + bp[k];
    __syncthreads();
    for (int off = 64; off > 0; off >>= 1) {
        if (k < off) red[k] += red[k + off];
        __syncthreads();
    }
    if (k == 0) konst[0] = (float)N_NODES * red[0];
}

// ---------------------------------------------------------------------------
// per-view: agg[n] = norm[n]*x[n]  (self term), then scatter norm[src]*x[src]
// ---------------------------------------------------------------------------
__global__ void k_scale(const float4* __restrict__ x4, const float* __restrict__ norm,
                        float4* __restrict__ agg4, int total) {
    int i = blockIdx.x * blockDim.x + threadIdx.x;
    if (i >= total) return;
    float nn = norm[i >> 5];           // 32 float4 chunks per row
    float4 val = x4[i];
    val.x *= nn; val.y *= nn; val.z *= nn; val.w *= nn;
    agg4[i] = val;
}

// one wave per edge: lane handles 4 floats (float4 gather + 4 f32 atomics)
__global__ void k_edge(const float* __restrict__ x, const float* __restrict__ norm,
                       const int* __restrict__ src, const int* __restrict__ dst,
                       float* __restrict__ agg, int E) {
    unsigned tid = blockIdx.x * blockDim.x + threadIdx.x;
    int e = tid >> 5;
    int lane = tid & 31;
    if (e >= E) return;
    int s = src[e], d = dst[e];
    float ns = norm[s];
    const float4 xv = *(const float4*)(x + (size_t)s * FD + lane * 4);
    float* out = agg + (size_t)d * FD + lane * 4;
    atomicAdd(out + 0, ns * xv.x);
    atomicAdd(out + 1, ns * xv.y);
    atomicAdd(out + 2, ns * xv.z);
    atomicAdd(out + 3, ns * xv.w);
}

// ---------------------------------------------------------------------------
// WMMA GEMM + fused relu / weighted reduction:
//   per node tile (16 rows): z = (agg*norm) @ W1 + b1 ; relu ;
//   partial += c[m] * z[m,n] * v[n] ; atomicAdd into view accumulator.
// Uses V_WMMA_F32_16X16X4_F32 (fp32 exact). One wave per 16-node tile,
// 8 column tiles x 32 K-steps = 256 WMMAs per wave. W1 staged in LDS as
// (k,k+1) float2 pairs -> one ds_load_b64 per B fragment.
// ---------------------------------------------------------------------------
__global__ __launch_bounds__(128) void k_gemm(
        const float* __restrict__ agg, const float* __restrict__ norm,
        const float* __restrict__ c, const float* __restrict__ W1,
        const float* __restrict__ b1, const float* __restrict__ v,
        float* __restrict__ sview_slot) {
    __shared__ float2 w1lds[64 * FD];   // 64KB: pair p holds {W1[2p][n], W1[2p+1][n]}

    const int tid = threadIdx.x;
    for (int idx = tid; idx < 64 * FD; idx += 128) {
        int p = idx >> 7, n = idx & 127;
        float2 t;
        t.x = W1[(p * 2)     * FD + n];
        t.y = W1[(p * 2 + 1) * FD + n];
        w1lds[idx] = t;
    }
    __syncthreads();

    const int wave = tid >> 5;
    const int lane = tid & 31;
    const int tile = blockIdx.x * 4 + wave;         // 6250 tiles total
    if (tile >= N_NODES / 16) return;               // wave-uniform exit

    const int node_base = tile * 16;
    const int n16   = lane & 15;                    // M for A, N for B/C
    const int khalf = lane >> 4;
    const int koff  = khalf * 2;

    const int row = node_base + n16;
    const float* arow = agg + (size_t)row * FD;
    const float nrm = norm[row];

    float b1v[8], vv[8];
#pragma unroll
    for (int t = 0; t < 8; ++t) {
        b1v[t] = b1[t * 16 + n16];
        vv[t]  = v [t * 16 + n16];
    }

    v8f acc[8];
#pragma unroll
    for (int t = 0; t < 8; ++t) acc[t] = (v8f){};

    for (int kb = 0; kb < FD; kb += 4) {
        // A fragment: {A[m][kb+koff], A[m][kb+koff+1]}, scaled by norm[m]
        const float2 af = *(const float2*)(arow + kb + koff);
        v2f a; a.x = af.x * nrm; a.y = af.y * nrm;
        const int prow = ((kb + koff) >> 1) << 7;   // pair index * 128
#pragma unroll
        for (int t = 0; t < 8; ++t) {
            float2 bf = w1lds[prow + t * 16 + n16]; // {W1[k][n], W1[k+1][n]}
            v2f b; b.x = bf.x; b.y = bf.y;
            acc[t] = __builtin_amdgcn_wmma_f32_16x16x4_f32(
                false, a, false, b, (short)0, acc[t], false, false);
        }
    }

    // epilogue: C/D layout -> element r of acc[t] is (M = r + khalf*8, N = t*16+n16)
    float partial = 0.0f;
    const int mb = node_base + khalf * 8;
#pragma unroll
    for (int r = 0; r < 8; ++r) {
        float cr = c[mb + r];
#pragma unroll
        for (int t = 0; t < 8; ++t) {
            float z = acc[t][r] + b1v[t];
            z = fmaxf(z, 0.0f);
            partial += cr * z * vv[t];
        }
    }
    for (int off = 16; off > 0; off >>= 1)
        partial += __shfl_down(partial, off, 32);
    if (lane == 0) atomicAdd(sview_slot, partial);
}

__global__ void k_final(const float* __restrict__ sview, const float* __restrict__ konst,
                        float* __restrict__ out) {
    float s1 = sview[0] + konst[0];
    float s2 = sview[1] + konst[0];
    float bce1 = fmaxf(s1, 0.0f) - s1 + log1pf(expf(-fabsf(s1)));  // label 1
    float bce2 = fmaxf(s2, 0.0f)      + log1pf(expf(-fabsf(s2)));  // label 0
    out[0] = 0.5f * (bce1 + bce2);
}

// ---------------------------------------------------------------------------
extern "C" void kernel_launch(void* const* d_in, const int* in_sizes, int n_in,
                              void* d_out, int out_size, void* d_ws, size_t ws_size,
                              hipStream_t stream) {
    const float* features = (const float*)d_in[0];
    const float* permuted = (const float*)d_in[1];
    const float* W1 = (const float*)d_in[2];
    const float* b1 = (const float*)d_in[3];
    const float* W2 = (const float*)d_in[4];
    const float* b2 = (const float*)d_in[5];
    const float* Wp = (const float*)d_in[6];
    const float* bp = (const float*)d_in[7];
    const int* src = (const int*)d_in[8];
    const int* dst = (const int*)d_in[9];

    // workspace layout (floats): agg[N*F] | norm[N] | c[N] | v[128] | sview[2] | konst[1]
    float* agg   = (float*)d_ws;
    float* norm  = agg + (size_t)N_NODES * FD;
    float* coeff = norm + N_NODES;
    float* v     = coeff + N_NODES;
    float* sview = v + FD;
    float* konst = sview + 2;

    const int N = N_NODES, E = N_EDGES;

    k_zero  <<<(N + 255) / 256, 256, 0, stream>>>(norm, coeff, sview, N);
    k_deg   <<<(E + 255) / 256, 256, 0, stream>>>(dst, norm, E);
    k_norm  <<<(N + 255) / 256, 256, 0, stream>>>(norm, N);
    k_srcsum<<<(E + 255) / 256, 256, 0, stream>>>(src, dst, norm, coeff, E);
    k_coeff <<<(N + 255) / 256, 256, 0, stream>>>(norm, coeff, N);
    k_proj  <<<1, 128, 0, stream>>>(W2, b2, Wp, bp, v, konst);

    for (int view = 0; view < 2; ++view) {
        const float* x = view ? permuted : features;
        k_scale<<<(N * 32 + 255) / 256, 256, 0, stream>>>(
            (const float4*)x, norm, (float4*)agg, N * 32);
        k_edge<<<(E * 32 + 255) / 256, 256, 0, stream>>>(x, norm, src, dst, agg, E);
        k_gemm<<<(N / 16 + 3) / 4, 128, 0, stream>>>(
            agg, norm, coeff, W1, b1, v, sview + view);
    }
    k_final<<<1, 1, 0, stream>>>(sview, konst, (float*)d_out);
}